// RNNSignatureModel_75144747810992
// MI455X (gfx1250) — compile-verified
//
#include <hip/hip_runtime.h>
#include <hip/hip_bf16.h>

#define BATCH   128
#define STREAMN 512
#define IN_CH   4
#define EXTRA   5
#define GROUPS  2
#define CCH     10          // C = IN_CH + 1 + EXTRA
#define WWIN    31          // W
#define STEPSZ  16          // STEP
#define NSTEPS  31          // LENGTH-1
#define SIGC    1110        // C + C^2 + C^3
#define FDIM    2220        // GROUPS*SIGC
#define RNN     256
#define G3      768         // 3*RNN
#define OUTD    10
#define MROWS   (WWIN*BATCH)   // 3968

typedef __attribute__((ext_vector_type(2))) float v2f;
typedef __attribute__((ext_vector_type(8))) float v8f;

// ---------------------------------------------------------------------------
// Kernel 1: path signatures (Chen recurrence, depth 3, C=10) per (bg, w) pair.
// One wave32 per pair; 4 waves per block. dx staged in LDS; S1 kept in LDS
// (replicated, allows dynamic-index reads); S2/S3 distributed by pair-index q.
// ---------------------------------------------------------------------------
__global__ void sig_kernel(const float* __restrict__ x,
                           const float* __restrict__ W_aug,
                           float* __restrict__ sig) {
    __shared__ float dxs[4][NSTEPS * CCH];
    __shared__ float s1s[4][CCH];
    const int wave = threadIdx.x >> 5;
    const int lane = threadIdx.x & 31;
    const int pair = blockIdx.x * 4 + wave;       // 0..7935
    const int bg = pair % (BATCH * GROUPS);
    const int w  = pair / (BATCH * GROUPS);
    const int b = bg / GROUPS, g = bg % GROUPS;

    // ---- build dx window (31 steps x 10 channels) into LDS ----
    const float dt = 1.0f / (float)(STREAMN - 1);
    for (int idx = lane; idx < NSTEPS * CCH; idx += 32) {
        const int j = idx / CCH, c = idx % CCH;
        const int s = w * STEPSZ + j;
        float d;
        if (c < IN_CH) {
            d = x[(b * STREAMN + s + 1) * IN_CH + c] - x[(b * STREAMN + s) * IN_CH + c];
        } else if (c == IN_CH) {
            d = dt;                                   // time channel; bias cancels in diff
        } else {
            const int e = c - (IN_CH + 1);
            float acc = 0.f;
            #pragma unroll
            for (int i = 0; i < IN_CH; ++i)
                acc += (x[(b * STREAMN + s + 1) * IN_CH + i] -
                        x[(b * STREAMN + s) * IN_CH + i]) * W_aug[(g * EXTRA + e) * IN_CH + i];
            d = acc;
        }
        dxs[wave][idx] = d;
    }
    if (lane < CCH) s1s[wave][lane] = 0.f;
    __syncthreads();

    // per-lane ownership: q = lane + 32*u (q < 100), indices a=q/10, bq=q%10
    int qa[4], qb[4];
    #pragma unroll
    for (int u = 0; u < 4; ++u) {
        const int q = lane + 32 * u;
        qa[u] = (q < 100) ? (q / 10) : 0;
        qb[u] = (q < 100) ? (q % 10) : 0;
    }

    float S2[4] = {0.f, 0.f, 0.f, 0.f};
    float S3[4][CCH];
    #pragma unroll
    for (int u = 0; u < 4; ++u)
        #pragma unroll
        for (int c = 0; c < CCH; ++c) S3[u][c] = 0.f;

    for (int j = 0; j < NSTEPS; ++j) {
        float dxv[CCH];
        #pragma unroll
        for (int c = 0; c < CCH; ++c) dxv[c] = dxs[wave][j * CCH + c];

        #pragma unroll
        for (int u = 0; u < 4; ++u) {
            const float dxa = dxs[wave][j * CCH + qa[u]];
            const float dxb = dxs[wave][j * CCH + qb[u]];
            const float s1a = s1s[wave][qa[u]];           // OLD S1
            const float p2q = 0.5f * dxa * dxb;           // P2[q]
            // newS3[q*10+c] += dx[c] * ( P2[q]/3 + S2old[q] + S1old[a]*0.5*dx[bq] )
            const float coef = p2q * (1.0f / 3.0f) + S2[u] + 0.5f * s1a * dxb;
            #pragma unroll
            for (int c = 0; c < CCH; ++c) S3[u][c] += coef * dxv[c];
            // newS2[q] = S2old[q] + P2[q] + S1old[a]*dx[bq]
            S2[u] += dxb * (0.5f * dxa + s1a);
        }
        __syncthreads();
        if (lane < CCH) s1s[wave][lane] += dxs[wave][j * CCH + lane];
        __syncthreads();
    }

    // ---- store signature: row = w*B + b, col offset g*SIGC; order [S1,S2,S3] ----
    float* dst = sig + (size_t)(w * BATCH + b) * FDIM + g * SIGC;
    if (lane == 0) {
        #pragma unroll
        for (int i = 0; i < CCH; ++i) dst[i] = s1s[wave][i];
    }
    #pragma unroll
    for (int u = 0; u < 4; ++u) {
        const int q = lane + 32 * u;
        if (q < 100) {
            dst[CCH + q] = S2[u];
            #pragma unroll
            for (int c = 0; c < CCH; ++c) dst[CCH + 100 + q * 10 + c] = S3[u][c];
        }
    }
}

// ---------------------------------------------------------------------------
// WMMA GEMM helper: one chunk of STEPS K-steps (4 K each). All operands for
// the chunk are loaded into distinct registers FIRST (clauses, deep loadcnt
// pipelining), then 4*STEPS WMMAs are issued. Avoids the one-register B
// operand recycling that forces s_wait_loadcnt 0 before every WMMA.
// ---------------------------------------------------------------------------
template <int STEPS>
__device__ __forceinline__ void wmma_chunk(const float* __restrict__ arow,
                                           const float* __restrict__ b0p,
                                           const float* __restrict__ b1p,
                                           const float* __restrict__ b2p,
                                           const float* __restrict__ b3p,
                                           int kk,
                                           v8f& c0, v8f& c1, v8f& c2, v8f& c3) {
    v2f a[STEPS], b0[STEPS], b1[STEPS], b2[STEPS], b3[STEPS];
    #pragma unroll
    for (int t = 0; t < STEPS; ++t) {
        a[t]  = *(const v2f*)(arow + kk + 4 * t);
        b0[t] = *(const v2f*)(b0p  + kk + 4 * t);
        b1[t] = *(const v2f*)(b1p  + kk + 4 * t);
        b2[t] = *(const v2f*)(b2p  + kk + 4 * t);
        b3[t] = *(const v2f*)(b3p  + kk + 4 * t);
    }
    #pragma unroll
    for (int t = 0; t < STEPS; ++t) {
        c0 = __builtin_amdgcn_wmma_f32_16x16x4_f32(false, a[t], false, b0[t], (short)0, c0, false, false);
        c1 = __builtin_amdgcn_wmma_f32_16x16x4_f32(false, a[t], false, b1[t], (short)0, c1, false, false);
        c2 = __builtin_amdgcn_wmma_f32_16x16x4_f32(false, a[t], false, b2[t], (short)0, c2, false, false);
        c3 = __builtin_amdgcn_wmma_f32_16x16x4_f32(false, a[t], false, b3[t], (short)0, c3, false, false);
    }
}

// ---------------------------------------------------------------------------
// Kernel 2: fp32 WMMA GEMM  C[M,N] = A[M,K] @ Wt[N,K]^T + bias[N]
// K compile-time; 16x64 tile per wave (4 N-subtiles share each A load).
// Block loop advances 5 base pointers once per 64 K; inside, 4 chunks of
// 4 K-steps (load-16, wmma-16) give immediate offsets + deep load pipelining.
// A layout per ISA: lanes 0-15 hold rows M=0..15 with (K=0,K=1); lanes 16-31
// the same rows with (K=2,K=3). N % 64 == 0, M % 16 == 0, K % 4 == 0.
// ---------------------------------------------------------------------------
template <int K>
__global__ void wmma_gemm_nt(const float* __restrict__ A,
                             const float* __restrict__ Wt,
                             const float* __restrict__ bias,
                             float* __restrict__ Cout,
                             int M, int N) {
    const int wave = threadIdx.x >> 5;
    const int lane = threadIdx.x & 31;
    const int tile = blockIdx.x * (blockDim.x >> 5) + wave;
    const int ntn  = N >> 6;                  // 64-wide N supertiles
    const int tm = tile / ntn, tn = tile % ntn;
    if (tm >= (M >> 4)) return;               // wave-uniform: EXEC stays all-ones

    const int half = lane >> 4, lr = lane & 15;
    const float* arow = A + (size_t)(tm * 16 + lr) * K + 2 * half;
    const float* b0p = Wt + (size_t)(tn * 64 +  0 + lr) * K + 2 * half;
    const float* b1p = Wt + (size_t)(tn * 64 + 16 + lr) * K + 2 * half;
    const float* b2p = Wt + (size_t)(tn * 64 + 32 + lr) * K + 2 * half;
    const float* b3p = Wt + (size_t)(tn * 64 + 48 + lr) * K + 2 * half;

    v8f c0 = {}, c1 = {}, c2 = {}, c3 = {};
    constexpr int KB   = 64;        // K elements per block (16 WMMA steps)
    constexpr int NBLK = K / KB;
    constexpr int KREM = K % KB;

    for (int blk = 0; blk < NBLK; ++blk) {
        if (blk + 1 < NBLK || KREM) {         // prefetch next K block
            __builtin_prefetch(arow + KB, 0, 3);
            __builtin_prefetch(b0p + KB, 0, 3);
            __builtin_prefetch(b1p + KB, 0, 3);
            __builtin_prefetch(b2p + KB, 0, 3);
            __builtin_prefetch(b3p + KB, 0, 3);
        }
        #pragma unroll
        for (int kk = 0; kk < KB; kk += 16)
            wmma_chunk<4>(arow, b0p, b1p, b2p, b3p, kk, c0, c1, c2, c3);
        arow += KB; b0p += KB; b1p += KB; b2p += KB; b3p += KB;
    }
    if constexpr (KREM > 0) {                 // compile-time tail (44 for K=2220)
        constexpr int T4 = KREM / 16;         // full 4-step chunks
        constexpr int TR = (KREM % 16) / 4;   // leftover steps
        #pragma unroll
        for (int i = 0; i < T4; ++i)
            wmma_chunk<4>(arow, b0p, b1p, b2p, b3p, i * 16, c0, c1, c2, c3);
        if constexpr (TR > 0)
            wmma_chunk<TR>(arow, b0p, b1p, b2p, b3p, T4 * 16, c0, c1, c2, c3);
    }

    #pragma unroll
    for (int s = 0; s < 4; ++s) {
        const v8f cc = (s == 0) ? c0 : (s == 1) ? c1 : (s == 2) ? c2 : c3;
        const int n = tn * 64 + s * 16 + lr;
        const float bv = bias[n];
        #pragma unroll
        for (int v = 0; v < 8; ++v) {
            const int m = tm * 16 + v + 8 * half;  // C layout: VGPR v -> rows v / v+8
            Cout[(size_t)m * N + n] = cc[v] + bv;
        }
    }
}

// ---------------------------------------------------------------------------
// Kernel 3: GRU gate update for one scan step. gi/gh already include biases.
// ---------------------------------------------------------------------------
__global__ void gru_gate(const float* __restrict__ gi,   // [B, 768] for this step
                         const float* __restrict__ gh,   // [B, 768] (h @ W_hh^T + b_hh)
                         float* __restrict__ h) {        // [B, RNN]
    const int idx = blockIdx.x * blockDim.x + threadIdx.x;
    if (idx >= BATCH * RNN) return;
    const int b = idx / RNN, j = idx % RNN;
    const float ir = gi[b * G3 + j],            hr = gh[b * G3 + j];
    const float iz = gi[b * G3 + RNN + j],      hz = gh[b * G3 + RNN + j];
    const float in_ = gi[b * G3 + 2 * RNN + j], hn = gh[b * G3 + 2 * RNN + j];
    const float r = 1.f / (1.f + __expf(-(ir + hr)));
    const float z = 1.f / (1.f + __expf(-(iz + hz)));
    const float n = tanhf(in_ + r * hn);
    h[idx] = (1.f - z) * n + z * h[idx];
}

// ---------------------------------------------------------------------------
// Kernel 4: output head  out = sigmoid(h @ W_out^T + b_out)
// ---------------------------------------------------------------------------
__global__ void out_head(const float* __restrict__ h,
                         const float* __restrict__ W_out,
                         const float* __restrict__ b_out,
                         float* __restrict__ out) {
    const int idx = blockIdx.x * blockDim.x + threadIdx.x;
    if (idx >= BATCH * OUTD) return;
    const int b = idx / OUTD, o = idx % OUTD;
    float acc = b_out[o];
    for (int j = 0; j < RNN; ++j) acc += h[b * RNN + j] * W_out[o * RNN + j];
    out[idx] = 1.f / (1.f + __expf(-acc));
}

extern "C" void kernel_launch(void* const* d_in, const int* in_sizes, int n_in,
                              void* d_out, int out_size, void* d_ws, size_t ws_size,
                              hipStream_t stream) {
    const float* x     = (const float*)d_in[0];
    const float* W_aug = (const float*)d_in[1];
    // d_in[2] = b_aug: cancels in the increment (dx), unused
    const float* W_ih  = (const float*)d_in[3];
    const float* W_hh  = (const float*)d_in[4];
    const float* b_ih  = (const float*)d_in[5];
    const float* b_hh  = (const float*)d_in[6];
    const float* W_out = (const float*)d_in[7];
    const float* b_out = (const float*)d_in[8];
    float* out = (float*)d_out;

    char* ws = (char*)d_ws;
    float* sig = (float*)ws;                                   // MROWS x FDIM
    float* gi  = (float*)(ws + (size_t)MROWS * FDIM * 4);      // MROWS x G3
    float* gh  = gi + (size_t)MROWS * G3;                      // BATCH x G3
    float* h   = gh + (size_t)BATCH * G3;                      // BATCH x RNN

    // 1) signatures: 7936 pairs, 4 waves/block
    sig_kernel<<<7936 / 4, 128, 0, stream>>>(x, W_aug, sig);

    // 2) all input gates in one GEMM: (3968 x 2220) @ (2220 x 768) + b_ih
    //    16x64 tiles: 248 * 12 = 2976 waves, 8 waves/block
    wmma_gemm_nt<FDIM><<<2976 / 8, 256, 0, stream>>>(sig, W_ih, b_ih, gi, MROWS, G3);

    // 3) GRU scan over 31 steps
    hipMemsetAsync(h, 0, (size_t)BATCH * RNN * 4, stream);
    for (int w = 0; w < WWIN; ++w) {
        // gh = h @ W_hh^T + b_hh : (128 x 256) @ (256 x 768); 8*12 = 96 waves
        wmma_gemm_nt<RNN><<<96 / 8, 256, 0, stream>>>(h, W_hh, b_hh, gh, BATCH, G3);
        gru_gate<<<(BATCH * RNN + 255) / 256, 256, 0, stream>>>(
            gi + (size_t)w * BATCH * G3, gh, h);
    }

    // 4) output head
    out_head<<<(BATCH * OUTD + 255) / 256, 256, 0, stream>>>(h, W_out, b_out, out);
}